// Decoder_61478161874954
// MI455X (gfx1250) — compile-verified
//
#include <hip/hip_runtime.h>
#include <math.h>

#define XS 256
#define NB 16
#define IMG_ELEMS (XS * XS)            /* 65536  */
#define BIMG_ELEMS (NB * IMG_ELEMS)    /* 1048576 */

#define TILE_M 64
#define TILE_N 64
#define TILE_K 32
#define NKB    (XS / TILE_K)  /* 8 K-panels */
#define A_STRIDE 36   /* 32 DW row + 4 DW TDM pad  -> 16 distinct banks for frag read */
#define B_STRIDE 72   /* 64 DW row + 8 DW TDM pad  -> half-wave K offsets in disjoint banks */

typedef float v2f __attribute__((ext_vector_type(2)));
typedef float v8f __attribute__((ext_vector_type(8)));
typedef unsigned int u32x4 __attribute__((ext_vector_type(4)));
typedef int i32x4 __attribute__((ext_vector_type(4)));
typedef int i32x8 __attribute__((ext_vector_type(8)));

// ---------------------------------------------------------------------------
// Build DFT basis: C[k][n] = cos(2*pi*k*n/256), S[k][n] = sin(2*pi*k*n/256)
// ---------------------------------------------------------------------------
__global__ void init_basis_kernel(float* __restrict__ Cm, float* __restrict__ Sm) {
    int t = blockIdx.x * blockDim.x + threadIdx.x;
    if (t >= IMG_ELEMS) return;
    int k = t >> 8;
    int n = t & 255;
    int ph = (k * n) & 255;
    float ang = (float)ph * 0.024543692606170258f;  // 2*pi/256
    float s, c;
    __sincosf(ang, &s, &c);
    Cm[t] = c;
    Sm[t] = s;
}

// ---------------------------------------------------------------------------
// Hermitian-extend half-spectrum ctf (B,256,129) -> full real filter H (B,256,256)
// ---------------------------------------------------------------------------
__global__ void mirror_ctf_kernel(const float* __restrict__ ctf, float* __restrict__ H) {
    int t = blockIdx.x * blockDim.x + threadIdx.x;
    if (t >= BIMG_ELEMS) return;
    int b  = t >> 16;
    int ky = (t >> 8) & 255;
    int kx = t & 255;
    float v;
    if (kx <= 128) {
        v = ctf[((size_t)b * 256 + ky) * 129 + kx];
    } else {
        int kym = (256 - ky) & 255;
        int kxm = 256 - kx;            // in [1,127]
        v = ctf[((size_t)b * 256 + kym) * 129 + kxm];
    }
    H[t] = v;
}

__global__ void zero_kernel(float* __restrict__ p, int n) {
    int t = blockIdx.x * blockDim.x + threadIdx.x;
    if (t < n) p[t] = 0.0f;
}

// ---------------------------------------------------------------------------
// Bilinear scatter-add of projected points. One thread = one (batch, point).
// ---------------------------------------------------------------------------
__global__ void splat_kernel(const float* __restrict__ alignment,
                             const float* __restrict__ shifts,
                             const float* __restrict__ coords,
                             const float* __restrict__ values,
                             float* __restrict__ img, int N) {
    long long t = (long long)blockIdx.x * blockDim.x + threadIdx.x;
    if (t >= (long long)NB * N) return;
    int b = (int)(t / N);
    int n = (int)(t - (long long)b * N);

    float x = coords[3 * n + 0];
    float y = coords[3 * n + 1];
    float z = coords[3 * n + 2];
    const float* al = alignment + 6 * b;
    float cx = x * al[0] + y * al[1] + z * al[2] - shifts[2 * b + 0] + 128.0f;
    float cy = x * al[3] + y * al[4] + z * al[5] - shifts[2 * b + 1] + 128.0f;

    float x0 = floorf(cx), y0 = floorf(cy);
    float fx = cx - x0,    fy = cy - y0;
    int ix0 = (int)x0, iy0 = (int)y0;
    int ix1 = ix0 + 1,  iy1 = iy0 + 1;
    ix0 = min(max(ix0, 0), 255);
    ix1 = min(max(ix1, 0), 255);
    iy0 = min(max(iy0, 0), 255);
    iy1 = min(max(iy1, 0), 255);

    float v = values[n];
    float w00 = v * (1.0f - fy) * (1.0f - fx);
    float w01 = v * (1.0f - fy) * fx;
    float w10 = v * fy * (1.0f - fx);
    float w11 = v * fy * fx;

    float* im = img + (size_t)b * IMG_ELEMS;
    atomicAdd(&im[iy0 * XS + ix0], w00);
    atomicAdd(&im[iy0 * XS + ix1], w01);
    atomicAdd(&im[iy1 * XS + ix0], w10);
    atomicAdd(&im[iy1 * XS + ix1], w11);
}

// ---------------------------------------------------------------------------
// 5x5 Gaussian (sigma=1), SAME with zero padding. Direct 25-tap.
// ---------------------------------------------------------------------------
__global__ void conv5_kernel(const float* __restrict__ in, float* __restrict__ out) {
    int t = blockIdx.x * blockDim.x + threadIdx.x;
    if (t >= BIMG_ELEMS) return;
    int b = t >> 16;
    int y = (t >> 8) & 255;
    int x = t & 255;
    const float w[5] = {0.054488685f, 0.244201347f, 0.402619958f, 0.244201347f, 0.054488685f};
    const float* im = in + (size_t)b * IMG_ELEMS;
    float acc = 0.0f;
#pragma unroll
    for (int dy = -2; dy <= 2; ++dy) {
        int yy = y + dy;
        if (yy < 0 || yy > 255) continue;
        float wy = w[dy + 2];
#pragma unroll
        for (int dx = -2; dx <= 2; ++dx) {
            int xx = x + dx;
            if (xx < 0 || xx > 255) continue;
            acc += wy * w[dx + 2] * im[yy * XS + xx];
        }
    }
    out[t] = acc;
}

// ---------------------------------------------------------------------------
// TDM helpers: raw LDS offset + 2D tensor_load_to_lds descriptor (D# per
// CDNA5 ISA). Hardware LDS padding implements the
// bank-padded panel strides (pad_interval/pad_amount codes).
// ---------------------------------------------------------------------------
__device__ __forceinline__ unsigned lds_off_u32(const void* p) {
    return (unsigned)(uintptr_t)(const __attribute__((address_space(3))) void*)p;
}

__device__ __forceinline__ void tdm_load_2d(unsigned lds_addr, const float* gptr,
                                            int tile_x, int tile_y,
                                            int pad_interval_code, int pad_amount_code) {
    unsigned long long ga = (unsigned long long)(uintptr_t)gptr;
    u32x4 g0;
    g0[0] = 1u;                                          // count=1, user mode
    g0[1] = lds_addr;                                    // lds_addr (bytes)
    g0[2] = (unsigned)(ga & 0xFFFFFFFFu);                // global_addr[31:0]
    g0[3] = (unsigned)((ga >> 32) & 0x01FFFFFFu)         // global_addr[56:32]
          | (2u << 30);                                  // type = 2 (image)
    i32x8 g1;
    g1[0] = (2 << 16)                                    // data_size = 4B
          | (1 << 20)                                    // pad_enable
          | (pad_interval_code << 22)
          | (pad_amount_code << 25);
    g1[1] = (int)((XS & 0xFFFFu) << 16);                 // tensor_dim0 lo16 (bar addr = 0)
    g1[2] = (int)((XS >> 16) | ((XS & 0xFFFFu) << 16));  // tensor_dim0 hi | tensor_dim1 lo
    g1[3] = (int)((XS >> 16) | ((unsigned)tile_x << 16));// tensor_dim1 hi | tile_dim0
    g1[4] = tile_y & 0xFFFF;                             // tile_dim1 (tile_dim2 = 0)
    g1[5] = XS;                                          // tensor_dim0_stride lo32
    g1[6] = 0;                                           // stride hi | dim1_stride lo
    g1[7] = 0;
    i32x4 z4 = {0, 0, 0, 0};
    i32x8 z8 = {0, 0, 0, 0, 0, 0, 0, 0};
    // 6-arg form (clang-23 / therock-10.0 headers): (g0, g1, g2, g3, g4, cpol)
    __builtin_amdgcn_tensor_load_to_lds(g0, g1, z4, z4, z8, 0);
}

// ---------------------------------------------------------------------------
// TDM double-buffered, LDS-tiled 256x256x256 GEMM per batch via
// V_WMMA_F32_16X16X4_F32:
//   Out[b] = alpha * ( s1 * A1[b] @ B1[b] + s2 * A2[b] @ B2[b] ) (* H[b] if given)
// Block = 128 threads (4 waves) -> 64x64 tile; wave w owns a 16x64 strip
// (4 v8f accumulators). Wave 0 issues TDM panel loads for iteration i+1 while
// all waves compute on panel i (TENSORcnt-ordered; s_wait_tensorcnt <= 4
// drains the current panel's 4 loads while the next 4 stay in flight).
// ---------------------------------------------------------------------------
__global__ __launch_bounds__(128)
void dft_mm_kernel(const float* __restrict__ A1, int a1s,
                   const float* __restrict__ B1, int b1s, float s1,
                   const float* __restrict__ A2, int a2s,
                   const float* __restrict__ B2, int b2s, float s2,
                   const float* __restrict__ mulp, float alpha,
                   float* __restrict__ out) {
    __shared__ __align__(16) float lA1[2][TILE_M * A_STRIDE];
    __shared__ __align__(16) float lB1[2][TILE_K * B_STRIDE];
    __shared__ __align__(16) float lA2[2][TILE_M * A_STRIDE];
    __shared__ __align__(16) float lB2[2][TILE_K * B_STRIDE];

    int tid  = threadIdx.x;       // 0..127
    int lane = tid & 31;          // wave32
    int wave = tid >> 5;          // 0..3
    int n0 = blockIdx.x * TILE_N;
    int m0 = blockIdx.y * TILE_M;
    int b  = blockIdx.z;

    int l15  = lane & 15;
    int hi   = lane >> 4;         // 0 or 1
    int kOff = hi * 2;

    const float* A1b = A1 + (size_t)b * (size_t)a1s;
    const float* B1b = B1 + (size_t)b * (size_t)b1s;
    const float* A2b = A2 ? (A2 + (size_t)b * (size_t)a2s) : nullptr;
    const float* B2b = B2 ? (B2 + (size_t)b * (size_t)b2s) : nullptr;
    bool dual = (A2b != nullptr);

    v8f acc[4];
#pragma unroll
    for (int c = 0; c < 4; ++c) acc[c] = (v8f){0.f,0.f,0.f,0.f,0.f,0.f,0.f,0.f};

    // Issue the 2 (or 4) TDM panel loads for K-panel kb into buffer bufi.
    // A panel: 32x64 tile, 32DW rows + 4DW pad  (codes 4,3)  -> stride 36.
    // B panel: 64x32 tile, 64DW rows + 8DW pad  (codes 5,7)  -> stride 72.
    auto issue_panels = [&](int bufi, int kb) {
        tdm_load_2d(lds_off_u32(&lA1[bufi][0]), A1b + m0 * XS + kb, TILE_K, TILE_M, 4, 3);
        tdm_load_2d(lds_off_u32(&lB1[bufi][0]), B1b + kb * XS + n0, TILE_N, TILE_K, 5, 7);
        if (dual) {
            tdm_load_2d(lds_off_u32(&lA2[bufi][0]), A2b + m0 * XS + kb, TILE_K, TILE_M, 4, 3);
            tdm_load_2d(lds_off_u32(&lB2[bufi][0]), B2b + kb * XS + n0, TILE_N, TILE_K, 5, 7);
        }
    };

    if (wave == 0) issue_panels(0, 0);

    for (int i = 0; i < NKB; ++i) {
        int cur = i & 1;
        if (wave == 0) {
            if (i + 1 < NKB) {
                issue_panels(cur ^ 1, (i + 1) * TILE_K);
                // TDM ops complete in order per wave: <=4 (or <=2) drains panel i.
                if (dual) __builtin_amdgcn_s_wait_tensorcnt(4);
                else      __builtin_amdgcn_s_wait_tensorcnt(2);
            } else {
                __builtin_amdgcn_s_wait_tensorcnt(0);
            }
        }
        __syncthreads();          // publish panel i to all waves

        int arow = (wave * 16 + l15) * A_STRIDE;
#pragma unroll
        for (int kk = 0; kk < TILE_K; kk += 4) {
            // A fragment (16x4 f32 layout): lanes 0-15 -> K=kk,kk+1 ; 16-31 -> kk+2,kk+3
            v2f a1 = *(const v2f*)&lA1[cur][arow + kk + kOff];
            a1.x *= s1; a1.y *= s1;
#pragma unroll
            for (int c = 0; c < 4; ++c) {
                v2f bf;
                bf.x = lB1[cur][(kk + kOff)     * B_STRIDE + c * 16 + l15];
                bf.y = lB1[cur][(kk + kOff + 1) * B_STRIDE + c * 16 + l15];
                acc[c] = __builtin_amdgcn_wmma_f32_16x16x4_f32(false, a1, false, bf,
                                                               (short)0, acc[c], false, false);
            }
            if (dual) {
                v2f a2 = *(const v2f*)&lA2[cur][arow + kk + kOff];
                a2.x *= s2; a2.y *= s2;
#pragma unroll
                for (int c = 0; c < 4; ++c) {
                    v2f bf;
                    bf.x = lB2[cur][(kk + kOff)     * B_STRIDE + c * 16 + l15];
                    bf.y = lB2[cur][(kk + kOff + 1) * B_STRIDE + c * 16 + l15];
                    acc[c] = __builtin_amdgcn_wmma_f32_16x16x4_f32(false, a2, false, bf,
                                                                   (short)0, acc[c], false, false);
                }
            }
        }

        __syncthreads();          // all waves done with panel i before it is re-filled
    }

    // ---- epilogue: C/D layout -> VGPR j is row (hi*8 + j), col l15 of subtile ----
    float* ob = out + (size_t)b * IMG_ELEMS;
    const float* mb = mulp ? (mulp + (size_t)b * IMG_ELEMS) : nullptr;
#pragma unroll
    for (int c = 0; c < 4; ++c) {
        int col = n0 + c * 16 + l15;
#pragma unroll
        for (int j = 0; j < 8; ++j) {
            int row = m0 + wave * 16 + hi * 8 + j;
            float v = alpha * acc[c][j];
            if (mb) v *= mb[row * XS + col];
            ob[row * XS + col] = v;
        }
    }
}

// ---------------------------------------------------------------------------
extern "C" void kernel_launch(void* const* d_in, const int* in_sizes, int n_in,
                              void* d_out, int out_size, void* d_ws, size_t ws_size,
                              hipStream_t stream) {
    const float* alignment = (const float*)d_in[0];  // (16, 6)
    const float* shifts    = (const float*)d_in[1];  // (16, 2)
    const float* coords    = (const float*)d_in[2];  // (N, 3)
    const float* values    = (const float*)d_in[3];  // (N,)
    const float* ctf       = (const float*)d_in[4];  // (16, 256, 129)
    int N = in_sizes[2] / 3;

    float* ws   = (float*)d_ws;
    float* Cm   = ws;                       // 64K floats
    float* Sm   = Cm + IMG_ELEMS;           // 64K floats
    float* H    = Sm + IMG_ELEMS;           // 1M floats
    float* buf0 = H + BIMG_ELEMS;           // 1M floats each
    float* buf1 = buf0 + BIMG_ELEMS;
    float* buf2 = buf1 + BIMG_ELEMS;
    float* buf3 = buf2 + BIMG_ELEMS;
    float* outp = (float*)d_out;

    init_basis_kernel<<<IMG_ELEMS / 256, 256, 0, stream>>>(Cm, Sm);
    mirror_ctf_kernel<<<BIMG_ELEMS / 256, 256, 0, stream>>>(ctf, H);
    zero_kernel<<<BIMG_ELEMS / 256, 256, 0, stream>>>(buf0, BIMG_ELEMS);

    long long total = (long long)NB * N;
    int blocks = (int)((total + 255) / 256);
    splat_kernel<<<blocks, 256, 0, stream>>>(alignment, shifts, coords, values, buf0, N);

    conv5_kernel<<<BIMG_ELEMS / 256, 256, 0, stream>>>(buf0, buf1);

    dim3 g(XS / TILE_N, XS / TILE_M, NB);   // (4,4,16)
    dim3 blk(128);
    const int BS = IMG_ELEMS;
    // Row FFT:  ReA = img@C  (buf2);  ImA = -img@S  (buf3)      [C,S symmetric]
    dft_mm_kernel<<<g, blk, 0, stream>>>(buf1, BS, Cm, 0,  1.f,
                                         nullptr, 0, nullptr, 0, 0.f,
                                         nullptr, 1.f, buf2);
    dft_mm_kernel<<<g, blk, 0, stream>>>(buf1, BS, Sm, 0, -1.f,
                                         nullptr, 0, nullptr, 0, 0.f,
                                         nullptr, 1.f, buf3);
    // Col FFT + filter:  ReG = H*(C@ReA + S@ImA) (buf0);  ImG = H*(C@ImA - S@ReA) (buf1)
    dft_mm_kernel<<<g, blk, 0, stream>>>(Cm, 0, buf2, BS,  1.f,
                                         Sm, 0, buf3, BS,  1.f,
                                         H, 1.f, buf0);
    dft_mm_kernel<<<g, blk, 0, stream>>>(Cm, 0, buf3, BS,  1.f,
                                         Sm, 0, buf2, BS, -1.f,
                                         H, 1.f, buf1);
    // Inverse col FFT:  ReA' = C@ReG - S@ImG (buf2);  ImA' = C@ImG + S@ReG (buf3)
    dft_mm_kernel<<<g, blk, 0, stream>>>(Cm, 0, buf0, BS,  1.f,
                                         Sm, 0, buf1, BS, -1.f,
                                         nullptr, 1.f, buf2);
    dft_mm_kernel<<<g, blk, 0, stream>>>(Cm, 0, buf1, BS,  1.f,
                                         Sm, 0, buf0, BS,  1.f,
                                         nullptr, 1.f, buf3);
    // Inverse row FFT (real part), fold 1/(256*256):
    //   out = (ReA'@C - ImA'@S) / 65536
    dft_mm_kernel<<<g, blk, 0, stream>>>(buf2, BS, Cm, 0,  1.f,
                                         buf3, BS, Sm, 0, -1.f,
                                         nullptr, 1.0f / 65536.0f, outp);
}